// KGATConv_38706245271755
// MI455X (gfx1250) — compile-verified
//
#include <hip/hip_runtime.h>
#include <math.h>

#define D 128
#define SLOPE 0.01f

typedef __attribute__((ext_vector_type(2))) float v2f;
typedef __attribute__((ext_vector_type(8))) float v8f;

#if defined(__gfx1250__) && __has_builtin(__builtin_amdgcn_wmma_f32_16x16x4_f32)
#define HAVE_WMMA_F32 1
#endif

// ---------------------------------------------------------------------------
// f32 16x16x4 WMMA wrapper. A: 16x4 (lane<16: M=lane, K pair {4k,4k+1};
// lane>=16: M=lane-16, K pair {4k+2,4k+3}). B: 4x16 mirrored (lane%16 = N).
// C/D: v8f, VGPR j holds M = 8*(lane>=16)+j, N = lane%16.
// ---------------------------------------------------------------------------
__device__ __forceinline__ v8f wmma_fallback(v2f a, v2f b, v8f c) {
  int lane = threadIdx.x & 31;
  int lp = lane & 15, hi = lane >> 4;
#pragma unroll
  for (int j = 0; j < 8; ++j) {
    int m = hi * 8 + j;
    float s = 0.f;
#pragma unroll
    for (int kk = 0; kk < 4; ++kk) {
      float av = __shfl((kk & 1) ? a.y : a.x, m + ((kk >> 1) << 4), 32);
      float bv = __shfl((kk & 1) ? b.y : b.x, lp + ((kk >> 1) << 4), 32);
      s += av * bv;
    }
    c[j] += s;
  }
  return c;
}

__device__ __forceinline__ v8f wmma4(v2f a, v2f b, v8f c) {
#ifdef HAVE_WMMA_F32
  return __builtin_amdgcn_wmma_f32_16x16x4_f32(false, a, false, b, (short)0, c,
                                               false, false);
#else
  return wmma_fallback(a, b, c);
#endif
}

// monotonic float -> uint ordering (for atomicMax on floats incl. negatives)
__device__ __forceinline__ unsigned ford(float f) {
  unsigned u = __float_as_uint(f);
  return (u & 0x80000000u) ? ~u : (u | 0x80000000u);
}
__device__ __forceinline__ float unford(unsigned o) {
  unsigned u = (o & 0x80000000u) ? (o & 0x7fffffffu) : ~o;
  return __uint_as_float(u);
}

// ---------------------------------------------------------------------------
// Edge binning by etype (counting sort, 16-aligned bins so tiles never mix r)
// ---------------------------------------------------------------------------
__global__ void k_hist(const int* __restrict__ ety, int* __restrict__ counts,
                       int E) {
  int e = blockIdx.x * blockDim.x + threadIdx.x;
  if (e < E) atomicAdd(&counts[ety[e]], 1);
}

__global__ void k_scan(const int* __restrict__ counts, int* __restrict__ aoff,
                       int R) {
  if (threadIdx.x == 0 && blockIdx.x == 0) {
    int off = 0;
    for (int r = 0; r < R; ++r) {
      aoff[r] = off;
      off += ((counts[r] + 15) >> 4) << 4;
    }
  }
}

__global__ void k_scatter(const int* __restrict__ ety,
                          const int* __restrict__ aoff, int* __restrict__ cursor,
                          int* __restrict__ sorted, int E) {
  int e = blockIdx.x * blockDim.x + threadIdx.x;
  if (e < E) {
    int r = ety[e];
    int p = atomicAdd(&cursor[r], 1);
    sorted[aoff[r] + p] = e;
  }
}

// ---------------------------------------------------------------------------
// Per 16-edge tile: t_r = h[src] @ W[r], h_r = h[dst] @ W[r] via WMMA,
// fused with att = sum_d t_r * tanh(h_r + efeat).
// Register-blocked: 16 v8f accumulators live in VGPRs; A/B streamed (each
// element loaded exactly once per wave). A loads are branchless: invalid
// (pad) lanes read row 0 and are zeroed by a 0/1 mask multiply.
// ---------------------------------------------------------------------------
__global__ void __launch_bounds__(256)
k_att(const float* __restrict__ h, const float* __restrict__ ef,
      const float* __restrict__ relw, const int* __restrict__ src,
      const int* __restrict__ dst, const int* __restrict__ ety,
      const int* __restrict__ sorted, float* __restrict__ att, int nTiles) {
  int lane = threadIdx.x & 31;
  int wid = threadIdx.x >> 5;
  int tile = blockIdx.x * (blockDim.x >> 5) + wid;
  if (tile >= nTiles) return;          // wave-uniform
  int base = tile * 16;
  int e0 = sorted[base];
  if (e0 < 0) return;                  // empty tile, wave-uniform
  int r = ety[e0];
  const float* W = relw + (size_t)r * D * D;

  int lp = lane & 15, hi = lane >> 4;
  int eid = sorted[base + lp];
  bool v = (eid >= 0);
  float mask = v ? 1.f : 0.f;
  long srow = v ? (long)src[eid] : 0;  // safe row for pad lanes
  long drow = v ? (long)dst[eid] : 0;
  const float* ps = h + srow * D;
  const float* pd = h + drow * D;

  v8f acc_t[8], acc_h[8];
  v8f z8 = {};
#pragma unroll
  for (int ct = 0; ct < 8; ++ct) {
    acc_t[ct] = z8;
    acc_h[ct] = z8;
  }

  for (int k = 0; k < 32; ++k) {
    int kb = 4 * k + 2 * hi;  // also the A column pair for this lane
    v2f at = *(const v2f*)(ps + kb) * mask;
    v2f ah = *(const v2f*)(pd + kb) * mask;
#pragma unroll
    for (int ct = 0; ct < 8; ++ct) {
      int n = ct * 16 + lp;
      v2f b;
      b.x = W[(size_t)kb * D + n];
      b.y = W[(size_t)(kb + 1) * D + n];
      acc_t[ct] = wmma4(at, b, acc_t[ct]);
      acc_h[ct] = wmma4(ah, b, acc_h[ct]);
    }
  }

  // epilogue: attp[j] = partial of sum_d t_r * tanh(h_r + efeat)
  int em[8];
#pragma unroll
  for (int j = 0; j < 8; ++j) em[j] = sorted[base + hi * 8 + j];

  float attp[8];
#pragma unroll
  for (int j = 0; j < 8; ++j) attp[j] = 0.f;

#pragma unroll
  for (int ct = 0; ct < 8; ++ct) {
    int n = ct * 16 + lp;
#pragma unroll
    for (int j = 0; j < 8; ++j) {
      if (em[j] >= 0) {
        float ev = ef[(size_t)em[j] * D + n];
        attp[j] += acc_t[ct][j] * tanhf(acc_h[ct][j] + ev);
      }
    }
  }
#pragma unroll
  for (int j = 0; j < 8; ++j) {
#pragma unroll
    for (int off = 8; off > 0; off >>= 1)
      attp[j] += __shfl_xor(attp[j], off, 16);
  }
  if (lp == 0) {
#pragma unroll
    for (int j = 0; j < 8; ++j)
      if (em[j] >= 0) att[em[j]] = attp[j];
  }
}

// ---------------------------------------------------------------------------
// Edge softmax over incoming edges of each dst node
// ---------------------------------------------------------------------------
__global__ void k_attmax(const float* __restrict__ att,
                         const int* __restrict__ dst,
                         unsigned* __restrict__ smax, int E) {
  int e = blockIdx.x * blockDim.x + threadIdx.x;
  if (e < E) atomicMax(&smax[dst[e]], ford(att[e]));
}

__global__ void k_expsum(float* __restrict__ att, const int* __restrict__ dst,
                         const unsigned* __restrict__ smax,
                         float* __restrict__ ssum, int E) {
  int e = blockIdx.x * blockDim.x + threadIdx.x;
  if (e < E) {
    int d = dst[e];
    float ex = __expf(att[e] - unford(smax[d]));
    att[e] = ex;
    atomicAdd(&ssum[d], ex);
  }
}

// message passing: h_nb[dst] += h[src] * a   (one thread per (edge, col))
__global__ void k_message(const float* __restrict__ h,
                          const float* __restrict__ att,
                          const float* __restrict__ ssum,
                          const int* __restrict__ src,
                          const int* __restrict__ dst,
                          float* __restrict__ h_nb, int E) {
  long t = (long)blockIdx.x * blockDim.x + threadIdx.x;
  int e = (int)(t >> 7);
  int dcol = (int)(t & 127);
  if (e >= E) return;
  int s = src[e], d = dst[e];
  float a = att[e] / ssum[d];
  atomicAdd(&h_nb[(size_t)d * D + dcol], h[(size_t)s * D + dcol] * a);
}

__global__ void k_transpose(const float* __restrict__ W1,
                            const float* __restrict__ W2,
                            float* __restrict__ WT1, float* __restrict__ WT2) {
  int i = blockIdx.x * blockDim.x + threadIdx.x;
  if (i < D * D) {
    int f = i / D, d = i % D;
    WT1[d * D + f] = W1[i];
    WT2[d * D + f] = W2[i];
  }
}

// ---------------------------------------------------------------------------
// out = leaky((h+h_nb) @ W_res^T) + leaky((h*h_nb) @ W_res2^T)
// Register-blocked the same way: 16 v8f accumulators, A/B streamed once,
// branchless edge-of-grid handling (clamped row + mask multiply).
// ---------------------------------------------------------------------------
__global__ void __launch_bounds__(256)
k_final(const float* __restrict__ h, const float* __restrict__ h_nb,
        const float* __restrict__ WT1, const float* __restrict__ WT2,
        float* __restrict__ out, int N) {
  int lane = threadIdx.x & 31;
  int wid = threadIdx.x >> 5;
  int tile = blockIdx.x * (blockDim.x >> 5) + wid;
  int nTiles = (N + 15) >> 4;
  if (tile >= nTiles) return;
  int lp = lane & 15, hi = lane >> 4;
  long rowRaw = (long)tile * 16 + lp;
  float mask = (rowRaw < N) ? 1.f : 0.f;
  long row = (rowRaw < N) ? rowRaw : (long)(N - 1);  // safe address
  const float* ph = h + row * D;
  const float* pn = h_nb + row * D;

  v8f acc1[8], acc2[8];
  v8f z8 = {};
#pragma unroll
  for (int ct = 0; ct < 8; ++ct) {
    acc1[ct] = z8;
    acc2[ct] = z8;
  }

  for (int k = 0; k < 32; ++k) {
    int kb = 4 * k + 2 * hi;
    v2f hv = *(const v2f*)(ph + kb) * mask;
    v2f nv = *(const v2f*)(pn + kb) * mask;
    v2f a1 = hv + nv;
    v2f a2 = hv * nv;
#pragma unroll
    for (int ct = 0; ct < 8; ++ct) {
      int n = ct * 16 + lp;
      v2f b1, b2;
      b1.x = WT1[(size_t)kb * D + n];
      b1.y = WT1[(size_t)(kb + 1) * D + n];
      b2.x = WT2[(size_t)kb * D + n];
      b2.y = WT2[(size_t)(kb + 1) * D + n];
      acc1[ct] = wmma4(a1, b1, acc1[ct]);
      acc2[ct] = wmma4(a2, b2, acc2[ct]);
    }
  }

#pragma unroll
  for (int ct = 0; ct < 8; ++ct) {
    int n = ct * 16 + lp;
#pragma unroll
    for (int j = 0; j < 8; ++j) {
      long orow = (long)tile * 16 + hi * 8 + j;
      if (orow < N) {
        float v1 = acc1[ct][j], v2 = acc2[ct][j];
        v1 = v1 > 0.f ? v1 : SLOPE * v1;
        v2 = v2 > 0.f ? v2 : SLOPE * v2;
        out[orow * D + n] = v1 + v2;
      }
    }
  }
}

// ---------------------------------------------------------------------------
extern "C" void kernel_launch(void* const* d_in, const int* in_sizes, int n_in,
                              void* d_out, int out_size, void* d_ws,
                              size_t ws_size, hipStream_t stream) {
  const float* nfeat = (const float*)d_in[0];
  const float* efeat = (const float*)d_in[1];
  const float* relw = (const float*)d_in[2];
  const float* W_res = (const float*)d_in[3];
  const float* W_res2 = (const float*)d_in[4];
  const int* src = (const int*)d_in[5];
  const int* dst = (const int*)d_in[6];
  const int* ety = (const int*)d_in[7];
  float* out = (float*)d_out;

  const int E = in_sizes[5];
  const int N = in_sizes[0] / D;
  const int R = in_sizes[2] / (D * D);
  const int S = ((E + 15) & ~15) + 16 * R;  // padded sorted-edge slots

  char* ws = (char*)d_ws;
  size_t off = 0;
  auto alloc = [&](size_t bytes) -> char* {
    char* p = ws + off;
    off += (bytes + 255) & ~(size_t)255;
    return p;
  };
  float* att = (float*)alloc((size_t)E * 4);
  unsigned* smax = (unsigned*)alloc((size_t)N * 4);
  float* ssum = (float*)alloc((size_t)N * 4);
  float* h_nb = (float*)alloc((size_t)N * D * 4);
  int* sorted = (int*)alloc((size_t)S * 4);
  int* counts = (int*)alloc((size_t)R * 4);
  int* cursor = (int*)alloc((size_t)R * 4);
  int* aoff = (int*)alloc((size_t)R * 4);
  float* WT1 = (float*)alloc((size_t)D * D * 4);
  float* WT2 = (float*)alloc((size_t)D * D * 4);
  (void)ws_size;
  (void)n_in;
  (void)out_size;

  hipMemsetAsync(smax, 0, (size_t)N * 4, stream);  // ord(x) > 0 for all reals
  hipMemsetAsync(ssum, 0, (size_t)N * 4, stream);
  hipMemsetAsync(h_nb, 0, (size_t)N * D * 4, stream);
  hipMemsetAsync(counts, 0, (size_t)R * 4, stream);
  hipMemsetAsync(cursor, 0, (size_t)R * 4, stream);
  hipMemsetAsync(sorted, 0xFF, (size_t)S * 4, stream);  // -1 sentinels

  k_hist<<<(E + 255) / 256, 256, 0, stream>>>(ety, counts, E);
  k_scan<<<1, 32, 0, stream>>>(counts, aoff, R);
  k_scatter<<<(E + 255) / 256, 256, 0, stream>>>(ety, aoff, cursor, sorted, E);

  int nTiles = S / 16;
  k_att<<<(nTiles + 7) / 8, 256, 0, stream>>>(nfeat, efeat, relw, src, dst, ety,
                                              sorted, att, nTiles);

  k_attmax<<<(E + 255) / 256, 256, 0, stream>>>(att, dst, smax, E);
  k_expsum<<<(E + 255) / 256, 256, 0, stream>>>(att, dst, smax, ssum, E);

  long tot = (long)E * D;
  k_message<<<(int)((tot + 255) / 256), 256, 0, stream>>>(nfeat, att, ssum, src,
                                                          dst, h_nb, E);

  k_transpose<<<(D * D + 255) / 256, 256, 0, stream>>>(W_res, W_res2, WT1, WT2);

  int fTiles = (N + 15) / 16;
  k_final<<<(fTiles + 7) / 8, 256, 0, stream>>>(nfeat, h_nb, WT1, WT2, out, N);
}